// GraphSAGELinkPredictor_18528488915295
// MI455X (gfx1250) — compile-verified
//
#include <hip/hip_runtime.h>

typedef __attribute__((ext_vector_type(2))) float v2f;
typedef __attribute__((ext_vector_type(8))) float v8f;

#define N_NODES 100000
#define N_EDGES 1600000
#define N_PAIRS 1000000
#define CH 32

// ---------------------------------------------------------------- utilities
__global__ void zero_f32(float* __restrict__ p, int n) {
    int i = blockIdx.x * blockDim.x + threadIdx.x;
    if (i < n) p[i] = 0.0f;
}

// ------------------------------------------------------------------ degrees
__global__ void deg_kernel(const int* __restrict__ dst, float* __restrict__ deg) {
    int e = blockIdx.x * blockDim.x + threadIdx.x;
    if (e < N_EDGES) atomicAdd(&deg[dst[e]], 1.0f);
}

// -------------------------------------------------------- edge scatter-add
// one lane per (edge, channel): lanes of a wave cover one edge's 32 channels
__global__ void scatter_kernel(const float* __restrict__ feat,
                               const int* __restrict__ src,
                               const int* __restrict__ dst,
                               float* __restrict__ agg) {
    int idx = blockIdx.x * blockDim.x + threadIdx.x;
    if (idx >= N_EDGES * CH) return;
    int e = idx >> 5;
    int c = idx & 31;
    atomicAdd(&agg[dst[e] * CH + c], feat[src[e] * CH + c]);
}

// --------------------------------------------------- WMMA fused SAGE linear
// out[16n tile] = (agg/deg) @ Wl + xin @ Wr + bias   (optionally ReLU)
// One wave handles 16 nodes; fp32 WMMA 16x16x4, K=32 in 8 steps, 2 N-halves.
template <bool RELU>
__global__ __launch_bounds__(256)
void sage_linear(const float* __restrict__ agg, const float* __restrict__ xin,
                 const float* __restrict__ deg,
                 const float* __restrict__ Wl, const float* __restrict__ Wr,
                 const float* __restrict__ bias, float* __restrict__ out) {
    __shared__ float sWl[CH * CH];
    __shared__ float sWr[CH * CH];
    __shared__ float sB[CH];

    int t = threadIdx.x;
#pragma unroll
    for (int i = 0; i < 4; ++i) {
        sWl[t + i * 256] = Wl[t + i * 256];
        sWr[t + i * 256] = Wr[t + i * 256];
    }
    if (t < CH) sB[t] = bias[t];
    __syncthreads();

    int lane = t & 31;
    int tile = blockIdx.x * 8 + (t >> 5);
    int base = tile * 16;
    if (base >= N_NODES) return;   // uniform per-wave; N_NODES % 16 == 0

    // A-matrix 16x4 f32 layout: lanes 0-15 M=0-15 K={k0,k0+1}; lanes 16-31 K={k0+2,k0+3}
    int m  = lane & 15;
    int kh = (lane >> 4) << 1;            // 0 or 2
    int node = base + m;
    float rd = 1.0f / fmaxf(deg[node], 1.0f);
    const float* arow = agg + node * CH;
    const float* xrow = xin + node * CH;

#pragma unroll
    for (int half = 0; half < 2; ++half) {
        int n0 = half * 16;
        v8f acc = {0.f, 0.f, 0.f, 0.f, 0.f, 0.f, 0.f, 0.f};
#pragma unroll
        for (int k0 = 0; k0 < CH; k0 += 4) {
            v2f a, b;
            // neighbor-aggregate path through Wl (mean applied on load)
            a.x = arow[k0 + kh]     * rd;
            a.y = arow[k0 + kh + 1] * rd;
            b.x = sWl[(k0 + kh)     * CH + n0 + m];
            b.y = sWl[(k0 + kh + 1) * CH + n0 + m];
            acc = __builtin_amdgcn_wmma_f32_16x16x4_f32(false, a, false, b,
                                                        (short)0, acc, false, false);
            // root path through Wr
            a.x = xrow[k0 + kh];
            a.y = xrow[k0 + kh + 1];
            b.x = sWr[(k0 + kh)     * CH + n0 + m];
            b.y = sWr[(k0 + kh + 1) * CH + n0 + m];
            acc = __builtin_amdgcn_wmma_f32_16x16x4_f32(false, a, false, b,
                                                        (short)0, acc, false, false);
        }
        // C/D layout: lanes 0-15 -> M = v, N = lane; lanes 16-31 -> M = v+8, N = lane-16
        float bn = sB[n0 + m];
        int rbase = base + ((lane >> 4) << 3);
#pragma unroll
        for (int v = 0; v < 8; ++v) {
            float val = acc[v] + bn;
            if (RELU) val = fmaxf(val, 0.0f);
            out[(rbase + v) * CH + n0 + m] = val;
        }
    }
}

// ---------------------------------------------------------- pair dot product
// 8 lanes per pair: lane j loads float4 channels [4j..4j+3] of both endpoints
__global__ void pair_dot(const float* __restrict__ z, const int* __restrict__ pairs,
                         float* __restrict__ outp) {
    int gid = blockIdx.x * blockDim.x + threadIdx.x;
    int p = gid >> 3;
    int j = gid & 7;
    if (p >= N_PAIRS) return;
    int a = pairs[2 * p];
    int b = pairs[2 * p + 1];
    const float4* za = (const float4*)(z + a * CH);
    const float4* zb = (const float4*)(z + b * CH);
    float4 va = za[j];
    float4 vb = zb[j];
    float s = va.x * vb.x + va.y * vb.y + va.z * vb.z + va.w * vb.w;
    s += __shfl_down(s, 4, 8);
    s += __shfl_down(s, 2, 8);
    s += __shfl_down(s, 1, 8);
    if (j == 0) outp[p] = s;
}

// ------------------------------------------------------------------- launch
extern "C" void kernel_launch(void* const* d_in, const int* in_sizes, int n_in,
                              void* d_out, int out_size, void* d_ws, size_t ws_size,
                              hipStream_t stream) {
    (void)in_sizes; (void)n_in; (void)out_size; (void)ws_size;

    const float* x   = (const float*)d_in[0];
    const int*   ei  = (const int*)d_in[1];    // [2, E] row-major
    const int*   prs = (const int*)d_in[2];    // [P, 2] row-major
    const float* Wl1 = (const float*)d_in[3];
    const float* Wr1 = (const float*)d_in[4];
    const float* b1  = (const float*)d_in[5];
    const float* Wl2 = (const float*)d_in[6];
    const float* Wr2 = (const float*)d_in[7];
    const float* b2  = (const float*)d_in[8];

    const int* src = ei;
    const int* dst = ei + N_EDGES;
    float* logits = (float*)d_out;

    // workspace partition (all fp32): deg | agg | h | z
    float* deg = (float*)d_ws;
    float* agg = deg + N_NODES;
    float* h   = agg + N_NODES * CH;
    float* z   = h   + N_NODES * CH;

    const int B = 256;
    int nzero = N_NODES * (CH + 1);
    zero_f32<<<(nzero + B - 1) / B, B, 0, stream>>>(deg, nzero);   // deg + agg

    deg_kernel<<<(N_EDGES + B - 1) / B, B, 0, stream>>>(dst, deg);

    // ---- layer 1
    scatter_kernel<<<(N_EDGES * CH + B - 1) / B, B, 0, stream>>>(x, src, dst, agg);
    int nblk = ((N_NODES / 16) + 7) / 8;
    sage_linear<true><<<nblk, B, 0, stream>>>(agg, x, deg, Wl1, Wr1, b1, h);

    // ---- layer 2
    nzero = N_NODES * CH;
    zero_f32<<<(nzero + B - 1) / B, B, 0, stream>>>(agg, nzero);
    scatter_kernel<<<(N_EDGES * CH + B - 1) / B, B, 0, stream>>>(h, src, dst, agg);
    sage_linear<false><<<nblk, B, 0, stream>>>(agg, h, deg, Wl2, Wr2, b2, z);

    // ---- link prediction
    pair_dot<<<(N_PAIRS * 8 + B - 1) / B, B, 0, stream>>>(z, prs, logits);
}